// GANLoss_89008902242786
// MI455X (gfx1250) — compile-verified
//
#include <hip/hip_runtime.h>

// f32 WMMA 16x16x4: A = 16x4 f32 (2 VGPRs/lane), B = 4x16 f32 (2 VGPRs/lane),
// C/D = 16x16 f32 (8 VGPRs/lane).
typedef __attribute__((ext_vector_type(2))) float v2f;
typedef __attribute__((ext_vector_type(8))) float v8f;

__global__ void GANLoss_zero_kernel(float* __restrict__ out) {
    if (threadIdx.x == 0 && blockIdx.x == 0) out[0] = 0.0f;
}

__global__ __launch_bounds__(256) void GANLoss_gather_reduce_kernel(
    const float* __restrict__ prob,
    const int*   __restrict__ target,
    const float* __restrict__ reward,
    float*       __restrict__ out,
    int n, long long C) {
    const int tid    = blockIdx.x * blockDim.x + threadIdx.x;
    const int stride = gridDim.x * blockDim.x;

    // Fused gather: one scattered load per row, FMA with reward.
    float acc = 0.0f;
    for (int i = tid; i < n; i += stride) {
        const long long t = (long long)target[i];
        const float p = prob[(long long)i * C + t];
        acc = fmaf(p, reward[i], acc);
    }
    // --- exact wave32 reduction via V_WMMA_F32_16X16X4_F32 -----------------
    // A[16x4]: lane L<16 holds {A[L][0], A[L][1]}, lane L>=16 holds
    // {A[L-16][2], A[L-16][3]}. Put this lane's partial in slot 0, zero the
    // other. B = all ones => D[m][n] = p_m + p_{m+16} for every column n.
    v2f a; a[0] = acc;  a[1] = 0.0f;
    v2f b; b[0] = 1.0f; b[1] = 1.0f;
    v8f c = {0.f, 0.f, 0.f, 0.f, 0.f, 0.f, 0.f, 0.f};
    c = __builtin_amdgcn_wmma_f32_16x16x4_f32(
        /*neg_a=*/false, a, /*neg_b=*/false, b,
        /*c_mod=*/(short)0, c, /*reuse_a=*/false, /*reuse_b=*/false);
    // Lane 0 accumulators hold D[0..7][0]; lane 16 holds D[8..15][0].
    float s = ((c[0] + c[1]) + (c[2] + c[3])) + ((c[4] + c[5]) + (c[6] + c[7]));
    float wave_sum = s + __shfl_xor(s, 16, 32);   // total of all 32 partials

    // --- block reduction (8 waves) -----------------------------------------
    __shared__ float wsum[8];
    const int lane = threadIdx.x & 31;
    const int wave = threadIdx.x >> 5;
    if (lane == 0) wsum[wave] = wave_sum;
    __syncthreads();
    if (threadIdx.x == 0) {
        float bsum = 0.0f;
        const int nw = blockDim.x >> 5;
        for (int w = 0; w < nw; ++w) bsum += wsum[w];
        atomicAdd(out, -bsum);                    // negate at the end
    }
}

extern "C" void kernel_launch(void* const* d_in, const int* in_sizes, int n_in,
                              void* d_out, int out_size, void* d_ws, size_t ws_size,
                              hipStream_t stream) {
    (void)n_in; (void)out_size; (void)d_ws; (void)ws_size;
    const float* prob   = (const float*)d_in[0];
    const int*   target = (const int*)d_in[1];
    const float* reward = (const float*)d_in[2];
    float*       out    = (float*)d_out;

    const long long total = (long long)in_sizes[0];   // N * C
    const int       n     = in_sizes[1];              // N (rows)
    const long long C     = total / (long long)n;     // classes per row

    GANLoss_zero_kernel<<<1, 1, 0, stream>>>(out);

    const int block = 256;
    int grid = (n + block - 1) / block;               // 8192/256 = 32 blocks
    if (grid < 1) grid = 1;
    GANLoss_gather_reduce_kernel<<<grid, block, 0, stream>>>(
        prob, target, reward, out, n, C);
}